// TransformerConv_12584254177711
// MI455X (gfx1250) — compile-verified
//
#include <hip/hip_runtime.h>
#include <math.h>

// TransformerConv for MI455X (gfx1250, wave32).
//   N=50000 nodes, E=800000 edges, D=128, H=4 heads, C=32 -> H*C=128.
// Pipeline:
//   K0: zero-init node_max/node_sum/node_feat(d_out)
//   K1: fused node-level GEMM  Q|K|V|S = x @ W* + b*   (fp32 WMMA 16x16x4,
//       one wave = 64 rows x 16 cols -> 4x B-fragment reuse)
//   K2: per-edge alpha = SCALE * <q[src],k[trg]>_head ; atomicMax seg-max
//   K3: per-(edge,head) exp(alpha - max) ; atomicAdd seg-sum
//   K4: per-edge scatter-add  alpha_norm * V[trg] -> node_feat (in d_out)
//   K5: per-node gated residual with wave32 shfl_xor reduction

#define D_IN   128
#define HC     128
#define NH     4
#define MT     4                       // M tiles per wave in the GEMM
#define SCALE_F 0.17677669529663687f   // 1/sqrt(32)

typedef float v2f __attribute__((ext_vector_type(2)));
typedef float v8f __attribute__((ext_vector_type(8)));

// order-preserving float <-> uint mapping for atomicMax on floats
__device__ __forceinline__ unsigned fenc(float f) {
  unsigned u = __float_as_uint(f);
  return (u & 0x80000000u) ? ~u : (u | 0x80000000u);
}
__device__ __forceinline__ float fdec(unsigned u) {
  unsigned b = (u & 0x80000000u) ? (u & 0x7fffffffu) : ~u;
  return __uint_as_float(b);
}

__device__ __forceinline__ void atomic_add_f32(float* p, float v) {
  __hip_atomic_fetch_add(p, v, __ATOMIC_RELAXED, __HIP_MEMORY_SCOPE_AGENT);
}

// ---------------------------------------------------------------- K0: init
__global__ __launch_bounds__(256)
void k_init(unsigned* __restrict__ node_max_u, float* __restrict__ node_sum,
            float* __restrict__ node_feat, int n4, int nfeat) {
  const int i = blockIdx.x * blockDim.x + threadIdx.x;
  if (i < n4) { node_max_u[i] = 0u; node_sum[i] = 0.0f; }   // fenc-domain -inf
  if (i < nfeat) node_feat[i] = 0.0f;
}

// ------------------------------------------------- K1: fused QKVS GEMM (WMMA)
// One wave -> 4 vertically stacked 16x16 tiles (64 rows x 16 cols) of
// Out = x[N,128] @ W[128,128] + b.  B fragments loaded once per K-step and
// reused for all 4 M tiles.  blockIdx.x: 64-row block, wave-in-block: column
// tile, blockIdx.y: which of the 4 weight matrices.
__global__ __launch_bounds__(256)
void k_gemm_qkvs(const float* __restrict__ x,
                 const float* __restrict__ Wq, const float* __restrict__ bq,
                 const float* __restrict__ Wk, const float* __restrict__ bk,
                 const float* __restrict__ Wv, const float* __restrict__ bv,
                 const float* __restrict__ Wsk, const float* __restrict__ bsk,
                 float* __restrict__ Q, float* __restrict__ K,
                 float* __restrict__ V, float* __restrict__ S, int n_nodes) {
  const int lane = threadIdx.x & 31;
  const int wv   = threadIdx.x >> 5;          // 0..7 -> column tile
  const int m0   = blockIdx.x * (16 * MT);
  const int n0   = wv * 16;

  const float* W; const float* b; float* O;
  switch (blockIdx.y) {
    case 0:  W = Wq;  b = bq;  O = Q; break;
    case 1:  W = Wk;  b = bk;  O = K; break;
    case 2:  W = Wv;  b = bv;  O = V; break;
    default: W = Wsk; b = bsk; O = S; break;
  }

  const int row  = lane & 15;                 // M (A) / N (B) within tile
  const int koff = (lane >> 4) << 1;          // lanes 16-31 hold K+2,K+3

  // bias: every row of the tile gets b[n0+col]; C/D col == lane&15
  v8f acc[MT];
  const float bias = b[n0 + row];
#pragma unroll
  for (int t = 0; t < MT; ++t)
#pragma unroll
    for (int i = 0; i < 8; ++i) acc[t][i] = bias;

  const float* ap[MT];
#pragma unroll
  for (int t = 0; t < MT; ++t) {
    int r = m0 + 16 * t + row;
    if (r >= n_nodes) r = n_nodes - 1;        // clamp (EXEC stays all-ones)
    ap[t] = x + (size_t)r * D_IN + koff;
  }
  const float* bp = W + (size_t)koff * HC + n0 + row;    // B: row-major W

#pragma unroll 2
  for (int k = 0; k < D_IN; k += 4) {
    v2f bf;
    bf.x = bp[(size_t)k * HC];                // B[k+koff][n]
    bf.y = bp[(size_t)(k + 1) * HC];          // B[k+koff+1][n]
#pragma unroll
    for (int t = 0; t < MT; ++t) {
      v2f a = *(const v2f*)(ap[t] + k);       // A[m][k+koff], A[m][k+koff+1]
      acc[t] = __builtin_amdgcn_wmma_f32_16x16x4_f32(
          /*neg_a=*/false, a, /*neg_b=*/false, bf,
          /*c_mod=*/(short)0, acc[t], /*reuse_a=*/false, /*reuse_b=*/false);
    }
  }

  // C/D layout: VGPR i -> M = i + 8*(lane>=16), N = lane&15
  const int rhalf = (lane >> 4) << 3;
  const int col   = n0 + row;
  if (m0 + 16 * MT <= n_nodes) {              // fast path: no bounds checks
#pragma unroll
    for (int t = 0; t < MT; ++t) {
      float* op = O + (size_t)(m0 + 16 * t + rhalf) * HC + col;
#pragma unroll
      for (int i = 0; i < 8; ++i) op[(size_t)i * HC] = acc[t][i];
    }
  } else {
#pragma unroll
    for (int t = 0; t < MT; ++t)
#pragma unroll
      for (int i = 0; i < 8; ++i) {
        const int r = m0 + 16 * t + rhalf + i;
        if (r < n_nodes) O[(size_t)r * HC + col] = acc[t][i];
      }
  }
}

// --------------------------------------- K2: per-edge scores + segment max
// One wave per edge; lane handles 4 channels; heads = groups of 8 lanes.
__global__ __launch_bounds__(256)
void k_edge_alpha(const float* __restrict__ Q, const float* __restrict__ Kf,
                  const int* __restrict__ src, const int* __restrict__ trg,
                  float* __restrict__ alpha_raw, unsigned* __restrict__ node_max_u,
                  int E) {
  const int e = (int)((blockIdx.x * (unsigned)blockDim.x + threadIdx.x) >> 5);
  if (e >= E) return;
  const int lane = threadIdx.x & 31;
  const int s = src[e];
  const int t = trg[e];
  const float4 q = *(const float4*)(Q  + (size_t)s * HC + lane * 4);
  const float4 k = *(const float4*)(Kf + (size_t)t * HC + lane * 4);
  float p = q.x * k.x + q.y * k.y + q.z * k.z + q.w * k.w;
  p += __shfl_xor(p, 1, 32);
  p += __shfl_xor(p, 2, 32);
  p += __shfl_xor(p, 4, 32);                  // sum within each 8-lane head
  if ((lane & 7) == 0) {
    const int h = lane >> 3;
    const float a = p * SCALE_F;
    alpha_raw[(size_t)e * NH + h] = a;
    atomicMax(node_max_u + (size_t)t * NH + h, fenc(a));
  }
}

// -------------------------------------------- K3: exp(alpha-max) + segment sum
__global__ __launch_bounds__(256)
void k_edge_exp(const float* __restrict__ alpha_raw, const int* __restrict__ trg,
                const unsigned* __restrict__ node_max_u,
                float* __restrict__ alpha_exp, float* __restrict__ node_sum, int E) {
  const int i = blockIdx.x * blockDim.x + threadIdx.x;
  if (i >= E * NH) return;
  const int e = i >> 2;
  const int h = i & 3;
  const int t = trg[e];
  const float m  = fdec(node_max_u[(size_t)t * NH + h]);
  const float ae = __expf(alpha_raw[i] - m);
  alpha_exp[i] = ae;
  atomic_add_f32(node_sum + (size_t)t * NH + h, ae);
}

// ------------------------------------- K4: weighted scatter-add of V into d_out
__global__ __launch_bounds__(256)
void k_edge_scatter(const float* __restrict__ V, const float* __restrict__ alpha_exp,
                    const float* __restrict__ node_sum, const int* __restrict__ trg,
                    float* __restrict__ node_feat, int E) {
  const int e = (int)((blockIdx.x * (unsigned)blockDim.x + threadIdx.x) >> 5);
  if (e >= E) return;
  const int lane = threadIdx.x & 31;
  const int t = trg[e];
  const int h = lane >> 3;
  const float w = alpha_exp[(size_t)e * NH + h] / node_sum[(size_t)t * NH + h];
  const float4 v = *(const float4*)(V + (size_t)t * HC + lane * 4);
  float* p = node_feat + (size_t)t * HC + lane * 4;
  atomic_add_f32(p + 0, w * v.x);
  atomic_add_f32(p + 1, w * v.y);
  atomic_add_f32(p + 2, w * v.z);
  atomic_add_f32(p + 3, w * v.w);
}

// ------------------------------------------------- K5: gated residual per node
__global__ __launch_bounds__(256)
void k_node_out(const float* __restrict__ S, const float* __restrict__ Wg,
                const float* __restrict__ bg, float* __restrict__ out, int Nn) {
  const int n = (int)((blockIdx.x * (unsigned)blockDim.x + threadIdx.x) >> 5);
  if (n >= Nn) return;
  const int lane = threadIdx.x & 31;
  const size_t base = (size_t)n * HC + lane * 4;
  const float4 nf = *(const float4*)(out + base);
  const float4 sk = *(const float4*)(S + base);
  const int j = lane * 4;
  const float4 wa = *(const float4*)(Wg + j);            // weight for node_feat
  const float4 wb = *(const float4*)(Wg + HC + j);       // weight for skip
  const float4 wc = *(const float4*)(Wg + 2 * HC + j);   // weight for skip-nf
  float g = nf.x * wa.x + nf.y * wa.y + nf.z * wa.z + nf.w * wa.w
          + sk.x * wb.x + sk.y * wb.y + sk.z * wb.z + sk.w * wb.w
          + (sk.x - nf.x) * wc.x + (sk.y - nf.y) * wc.y
          + (sk.z - nf.z) * wc.z + (sk.w - nf.w) * wc.w;
  g += __shfl_xor(g, 1, 32);
  g += __shfl_xor(g, 2, 32);
  g += __shfl_xor(g, 4, 32);
  g += __shfl_xor(g, 8, 32);
  g += __shfl_xor(g, 16, 32);                 // butterfly: all lanes hold total
  g = 1.0f / (1.0f + __expf(-(g + bg[0])));
  float4 o;
  o.x = g * sk.x + (1.0f - g) * nf.x;
  o.y = g * sk.y + (1.0f - g) * nf.y;
  o.z = g * sk.z + (1.0f - g) * nf.z;
  o.w = g * sk.w + (1.0f - g) * nf.w;
  *(float4*)(out + base) = o;
}

// ---------------------------------------------------------------- launcher
extern "C" void kernel_launch(void* const* d_in, const int* in_sizes, int n_in,
                              void* d_out, int out_size, void* d_ws, size_t ws_size,
                              hipStream_t stream) {
  const float* x   = (const float*)d_in[0];
  const int*   ei  = (const int*)d_in[1];
  const float* Wq  = (const float*)d_in[2];
  const float* bq  = (const float*)d_in[3];
  const float* Wk  = (const float*)d_in[4];
  const float* bk  = (const float*)d_in[5];
  const float* Wv  = (const float*)d_in[6];
  const float* bv  = (const float*)d_in[7];
  const float* Wsk = (const float*)d_in[8];
  const float* bsk = (const float*)d_in[9];
  const float* Wg  = (const float*)d_in[10];
  const float* bg  = (const float*)d_in[11];

  const int Nn = in_sizes[0] / D_IN;
  const int E  = in_sizes[1] / 2;
  const int* src = ei;
  const int* trg = ei + E;

  float* ws = (float*)d_ws;
  size_t o = 0;
  float* Q = ws + o;         o += (size_t)Nn * HC;
  float* K = ws + o;         o += (size_t)Nn * HC;
  float* V = ws + o;         o += (size_t)Nn * HC;
  float* S = ws + o;         o += (size_t)Nn * HC;
  float* alpha_raw = ws + o; o += (size_t)E * NH;
  float* alpha_exp = ws + o; o += (size_t)E * NH;
  float* node_sum  = ws + o; o += (size_t)Nn * NH;
  unsigned* node_max_u = (unsigned*)(ws + o);

  float* node_feat = (float*)d_out;   // accumulate node_feat directly in d_out

  const int initN = Nn * HC;
  k_init<<<(initN + 255) / 256, 256, 0, stream>>>(node_max_u, node_sum, node_feat,
                                                  Nn * NH, initN);
  k_gemm_qkvs<<<dim3((Nn + 16 * MT - 1) / (16 * MT), 4), 256, 0, stream>>>(
      x, Wq, bq, Wk, bk, Wv, bv, Wsk, bsk, Q, K, V, S, Nn);
  const int eb = (E + 7) / 8;   // one wave per edge, 8 waves/block
  k_edge_alpha<<<eb, 256, 0, stream>>>(Q, K, src, trg, alpha_raw, node_max_u, E);
  k_edge_exp<<<(E * NH + 255) / 256, 256, 0, stream>>>(alpha_raw, trg, node_max_u,
                                                       alpha_exp, node_sum, E);
  k_edge_scatter<<<eb, 256, 0, stream>>>(V, alpha_exp, node_sum, trg, node_feat, E);
  k_node_out<<<(Nn + 7) / 8, 256, 0, stream>>>(S, Wg, bg, node_feat, Nn);
}